// LinearNode_26663156974249
// MI455X (gfx1250) — compile-verified
//
#include <hip/hip_runtime.h>
#include <stdint.h>

// LIF (LinearNode) multi-step forward: v = a*v + b*x_t; spike = (v>=1); v = spike?0:v
// T=64 sequential steps carried in registers; (N*D)=1M independent lanes.
// Memory-bound: 512 MB total traffic -> ~22us floor @ 23.3 TB/s.
// CDNA5 path: depth-8 async global->LDS prefetch pipeline (ASYNCcnt) + NT stores.
// Little's law: ~20KB+/SIMD must be in flight; 8 waves/SIMD x 8 x 512B = 32KB.

typedef __attribute__((ext_vector_type(4))) float float4v;
typedef __attribute__((ext_vector_type(4))) int   v4i;

#define BLOCK 256
#define DEPTH 8   // async pipeline depth (timesteps in flight); T must be >= DEPTH

#if defined(__gfx1250__) && __has_builtin(__builtin_amdgcn_global_load_async_to_lds_b128)
#define LN_ASYNC 1
#else
#define LN_ASYNC 0
#endif

#if __has_builtin(__builtin_amdgcn_s_wait_asynccnt)
#define WAIT_ASYNC(n) __builtin_amdgcn_s_wait_asynccnt(n)
#else
#define WAIT_ASYNC(n) asm volatile("s_wait_asynccnt " #n ::: "memory")
#endif

// Builtin prototype (probe-confirmed via diagnostics):
//   void __builtin_amdgcn_global_load_async_to_lds_b128(
//       v4i addrspace(1)* gsrc, v4i addrspace(3)* ldst, int offset, int cpol);
typedef __attribute__((address_space(1))) v4i as1_v4i;
typedef __attribute__((address_space(3))) v4i as3_v4i;

// One scan step on 4 packed elements. v updated in place, returns spikes.
__device__ __forceinline__ float4v ln_step(float4v& v, float4v xv, float a, float b) {
    float4v sp;
#pragma unroll
    for (int i = 0; i < 4; ++i) {
        float vv = fmaf(b, xv[i], a * v[i]);   // a*v + b*x
        sp[i]    = (vv >= 1.0f) ? 1.0f : 0.0f; // heaviside(v - 1)
        v[i]     = (vv >= 1.0f) ? 0.0f : vv;   // hard reset to 0
    }
    return sp;
}

__global__ __launch_bounds__(BLOCK) void linear_node_scan(
    const float* __restrict__ x, const float* __restrict__ a_p,
    const float* __restrict__ b_p, float* __restrict__ out,
    int stride_v4, int T)
{
    const int col = blockIdx.x * BLOCK + threadIdx.x;   // float4 column in (N*D)/4
    if (col >= stride_v4) return;

    const float a = a_p[0];
    const float b = b_p[0];

    const float4v* __restrict__ xg = (const float4v*)x;
    float4v*       __restrict__ og = (float4v*)out;

    float4v v = {0.0f, 0.0f, 0.0f, 0.0f};   // v_reset = 0

#if LN_ASYNC
    // Per-thread ring of DEPTH x 16B slots in LDS: 32KB per 256-thread block.
    __shared__ __align__(16) float lbuf[DEPTH][BLOCK * 4];

    // Prologue: fill the pipe with t = 0..DEPTH-1.
#pragma unroll
    for (int i = 0; i < DEPTH; ++i) {
        __builtin_amdgcn_global_load_async_to_lds_b128(
            (as1_v4i*)(xg + (size_t)i * stride_v4 + col),
            (as3_v4i*)&lbuf[i][threadIdx.x * 4], 0, 0);
    }

    // Steady state: exactly DEPTH loads outstanding at each wait, so
    // asynccnt <= DEPTH-1 guarantees (in-order completion) that t's tile landed.
    for (int t = 0; t < T - DEPTH; ++t) {
        WAIT_ASYNC(7);  // DEPTH-1

        const float* cur = &lbuf[t & (DEPTH - 1)][threadIdx.x * 4];
        float4v xv = *(const float4v*)cur;            // ds_load_b128 (own slot)
        float4v sp = ln_step(v, xv, a, b);
        __builtin_nontemporal_store(sp, og + (size_t)t * stride_v4 + col);

        // Keep the aliasing LDS read above from sinking below the refill.
        asm volatile("" ::: "memory");

        // Refill the slot just consumed with timestep t + DEPTH.
        __builtin_amdgcn_global_load_async_to_lds_b128(
            (as1_v4i*)(xg + (size_t)(t + DEPTH) * stride_v4 + col),
            (as3_v4i*)&lbuf[t & (DEPTH - 1)][threadIdx.x * 4], 0, 0);
    }

    // Drain: everything for the last DEPTH timesteps is (or becomes) resident.
    WAIT_ASYNC(0);
#pragma unroll
    for (int i = 0; i < DEPTH; ++i) {
        const int t = T - DEPTH + i;
        const float* cur = &lbuf[t & (DEPTH - 1)][threadIdx.x * 4];
        float4v xv = *(const float4v*)cur;
        float4v sp = ln_step(v, xv, a, b);
        __builtin_nontemporal_store(sp, og + (size_t)t * stride_v4 + col);
    }
#else
    // Fallback: direct streaming loads (non-temporal), serial t-loop.
    for (int t = 0; t < T; ++t) {
        float4v xv = __builtin_nontemporal_load(xg + (size_t)t * stride_v4 + col);
        float4v sp = ln_step(v, xv, a, b);
        __builtin_nontemporal_store(sp, og + (size_t)t * stride_v4 + col);
    }
#endif
}

extern "C" void kernel_launch(void* const* d_in, const int* in_sizes, int n_in,
                              void* d_out, int out_size, void* d_ws, size_t ws_size,
                              hipStream_t stream) {
    const float* x = (const float*)d_in[0];   // (T, N, D) fp32
    const float* a = (const float*)d_in[1];   // scalar
    const float* b = (const float*)d_in[2];   // scalar

    float* out = (float*)d_out;               // (T, N, D) fp32 spikes

    const int T = 64;
    const long long total = (long long)in_sizes[0];     // T*N*D = 67,108,864
    const int stride_v4 = (int)(total / T / 4);         // (N*D)/4 = 262,144

    const int nblocks = (stride_v4 + BLOCK - 1) / BLOCK; // 1024 blocks x 8 waves
    linear_node_scan<<<nblocks, BLOCK, 0, stream>>>(x, a, b, out, stride_v4, T);
}